// LabelSmoothing_60928406061363
// MI455X (gfx1250) — compile-verified
//
#include <hip/hip_runtime.h>

#define B_TOTAL 32768
#define C_DIM   2048
#define SMOOTH  0.05f
#define CONF    0.95f
#define NBLK    4096
#define NTHR    256
#define L2E     1.4426950408889634f
#define LN2     0.6931471805599453f

// Issue async global->LDS copy of one 8KB row slice handled by this thread:
// each of 256 threads moves 16B at [tid*16] and 16B at [tid*16 + 4096].
// Per CDNA5 ISA the instruction offset is added to both the LDS and memory address,
// so the immediate offset applies to both sides.
__device__ __forceinline__ void async_copy_row(const float* grow, unsigned ldsoff, int tid) {
    unsigned long long ga = (unsigned long long)(uintptr_t)grow + (unsigned)tid * 16u;
    unsigned lo = ldsoff + (unsigned)tid * 16u;
    asm volatile("global_load_async_to_lds_b128 %0, %1, off"
                 :: "v"(lo), "v"(ga) : "memory");
    asm volatile("global_load_async_to_lds_b128 %0, %1, off offset:4096"
                 :: "v"(lo), "v"(ga) : "memory");
}

__global__ __launch_bounds__(NTHR) void ls_main(const float* __restrict__ x,
                                                const int* __restrict__ target,
                                                const float* __restrict__ clw,
                                                float* __restrict__ partials)
{
    __shared__ __align__(16) float buf[2][C_DIM];   // double-buffered row staging (16 KB)
    __shared__ float red0[8], red1[8], red2[8];

    const int tid  = threadIdx.x;
    const int lane = tid & 31;
    const int wid  = tid >> 5;

    const unsigned lds0 = (unsigned)(uintptr_t)(&buf[0][0]);
    const unsigned lds1 = (unsigned)(uintptr_t)(&buf[1][0]);

    float loss_acc = 0.0f;
    float w_acc    = 0.0f;

    const int row0 = blockIdx.x;

    // Prologue: stage first row into buffer 0.
    async_copy_row(x + (size_t)row0 * C_DIM, lds0, tid);

    int cur = 0;
    for (int row = row0; row < B_TOTAL; row += NBLK) {
        // Prefetch next row into the other buffer (clamped redundant fetch on last iter).
        int nrow = row + NBLK;
        const float* gnext = x + (size_t)((nrow < B_TOTAL) ? nrow : row) * C_DIM;
        async_copy_row(gnext, cur ? lds0 : lds1, tid);

        // Wait for *current* row's 2 async ops (2 newer ones may stay in flight),
        // then publish the buffer block-wide.
        asm volatile("s_wait_asynccnt 2" ::: "memory");
        __syncthreads();

        const float* r = &buf[cur][0];
        const float4 a = ((const float4*)r)[tid * 2];
        const float4 b = ((const float4*)r)[tid * 2 + 1];

        // ---- pass 1: row max ----
        float m = fmaxf(fmaxf(fmaxf(a.x, a.y), fmaxf(a.z, a.w)),
                        fmaxf(fmaxf(b.x, b.y), fmaxf(b.z, b.w)));
        #pragma unroll
        for (int off = 16; off > 0; off >>= 1)
            m = fmaxf(m, __shfl_xor(m, off, 32));
        if (lane == 0) red0[wid] = m;
        __syncthreads();
        m = red0[0];
        #pragma unroll
        for (int j = 1; j < 8; ++j) m = fmaxf(m, red0[j]);

        // ---- pass 2: sum(exp(x-m)) and sum(x), both from LDS-resident registers ----
        float se = exp2f((a.x - m) * L2E) + exp2f((a.y - m) * L2E)
                 + exp2f((a.z - m) * L2E) + exp2f((a.w - m) * L2E)
                 + exp2f((b.x - m) * L2E) + exp2f((b.y - m) * L2E)
                 + exp2f((b.z - m) * L2E) + exp2f((b.w - m) * L2E);
        float sx = (a.x + a.y + a.z + a.w) + (b.x + b.y + b.z + b.w);
        #pragma unroll
        for (int off = 16; off > 0; off >>= 1) {
            se += __shfl_xor(se, off, 32);
            sx += __shfl_xor(sx, off, 32);
        }
        if (lane == 0) { red1[wid] = se; red2[wid] = sx; }
        __syncthreads();

        if (tid == 0) {
            float SE = 0.0f, SX = 0.0f;
            #pragma unroll
            for (int j = 0; j < 8; ++j) { SE += red1[j]; SX += red2[j]; }
            const float lse = m + LN2 * log2f(SE);
            const int   t   = target[row];
            const float w   = clw[t];
            const float xt  = buf[cur][t];          // gather straight from LDS
            loss_acc += CONF * (lse - xt) * w + SMOOTH * (lse - SX * (1.0f / C_DIM));
            w_acc    += w;
        }
        __syncthreads();   // everyone done reading buf[cur] before it is re-staged
        cur ^= 1;
    }

    if (tid == 0) {
        partials[2 * blockIdx.x]     = loss_acc;
        partials[2 * blockIdx.x + 1] = w_acc;
    }
}

__global__ __launch_bounds__(256) void ls_final(const float* __restrict__ partials,
                                                float* __restrict__ out)
{
    __shared__ float r0[8], r1[8];
    const int tid  = threadIdx.x;
    const int lane = tid & 31;
    const int wid  = tid >> 5;

    float ls = 0.0f, ws = 0.0f;
    #pragma unroll
    for (int i = tid; i < NBLK; i += 256) {
        ls += partials[2 * i];
        ws += partials[2 * i + 1];
    }
    #pragma unroll
    for (int off = 16; off > 0; off >>= 1) {
        ls += __shfl_xor(ls, off, 32);
        ws += __shfl_xor(ws, off, 32);
    }
    if (lane == 0) { r0[wid] = ls; r1[wid] = ws; }
    __syncthreads();
    if (tid == 0) {
        float L = 0.0f, W = 0.0f;
        #pragma unroll
        for (int j = 0; j < 8; ++j) { L += r0[j]; W += r1[j]; }
        out[0] = L / W;
    }
}

extern "C" void kernel_launch(void* const* d_in, const int* in_sizes, int n_in,
                              void* d_out, int out_size, void* d_ws, size_t ws_size,
                              hipStream_t stream) {
    const float* x      = (const float*)d_in[0];
    const int*   target = (const int*)d_in[1];
    const float* clw    = (const float*)d_in[2];
    float* partials = (float*)d_ws;            // 4096 * 2 floats = 32 KB
    float* out      = (float*)d_out;

    ls_main<<<NBLK, NTHR, 0, stream>>>(x, target, clw, partials);
    ls_final<<<1, 256, 0, stream>>>(partials, out);
}